// Discriminator_7533372637744
// MI455X (gfx1250) — compile-verified
//
#include <hip/hip_runtime.h>
#include <math.h>

#define N_NODES 512
#define BATCH   256
#define NTOT    (N_NODES * BATCH)   // 131072
#define FEAT    32
#define HID     64
#define NEDGE   2097152

typedef __attribute__((ext_vector_type(2))) float v2f;
typedef __attribute__((ext_vector_type(8))) float v8f;

// ---------------------------------------------------------------------------
// Degree / normalization kernels
// ---------------------------------------------------------------------------
__global__ void k_fill_ones(float* __restrict__ d) {
    int i = blockIdx.x * blockDim.x + threadIdx.x;
    d[i] = 1.0f;                       // self-loop contributes 1 to every degree
}

__global__ void k_deg_count(const int* __restrict__ dst, float* deg) {
    int e = blockIdx.x * blockDim.x + threadIdx.x;
    atomicAdd(&deg[dst[e]], 1.0f);     // global_atomic_add_f32, L2-resident
}

__global__ void k_deg_rsqrt(float* __restrict__ d) {
    int i = blockIdx.x * blockDim.x + threadIdx.x;
    d[i] = rsqrtf(d[i]);               // deg >= 1 always (self loop)
}

// ---------------------------------------------------------------------------
// WMMA GEMM: Y[N,64] = X[N,K] @ W[K,64]   (f32, V_WMMA_F32_16X16X4_F32)
// One wave handles a 16-row x 64-col tile (4 accumulators of v8f).
// A layout (16x4 f32): lanes 0-15 -> {K=0,K=1}, lanes 16-31 -> {K=2,K=3}
// B layout (4x16 f32): VGPR0 = row (k+2*half), VGPR1 = row (k+2*half+1), N=lane&15
// C/D layout: VGPR j -> M = j + 8*(lane>>4), N = lane&15
// Safe in place: every store data-depends on every A load (acc chain).
// ---------------------------------------------------------------------------
template <int K>
__global__ void k_gemm_wmma(const float* X, const float* __restrict__ W, float* Y) {
    const int lane = threadIdx.x & 31;
    const int wave = threadIdx.x >> 5;
    const int half = lane >> 4;        // 0 or 1
    const int r    = lane & 15;
    const long row0 = ((long)blockIdx.x * (blockDim.x >> 5) + wave) * 16;

    v8f acc0 = {}, acc1 = {}, acc2 = {}, acc3 = {};
    const float* xrow = X + (row0 + r) * K;

#pragma unroll
    for (int k = 0; k < K; k += 4) {
        v2f a;
        a.x = xrow[k + 2 * half];
        a.y = xrow[k + 2 * half + 1];
        const float* w0 = W + (k + 2 * half) * HID;   // W row (k+2*half)
        const float* w1 = w0 + HID;                   // W row (k+2*half+1)

        v2f b;
        b.x = w0[r];        b.y = w1[r];
        acc0 = __builtin_amdgcn_wmma_f32_16x16x4_f32(false, a, false, b, (short)0, acc0, false, false);
        b.x = w0[16 + r];   b.y = w1[16 + r];
        acc1 = __builtin_amdgcn_wmma_f32_16x16x4_f32(false, a, false, b, (short)0, acc1, false, false);
        b.x = w0[32 + r];   b.y = w1[32 + r];
        acc2 = __builtin_amdgcn_wmma_f32_16x16x4_f32(false, a, false, b, (short)0, acc2, false, false);
        b.x = w0[48 + r];   b.y = w1[48 + r];
        acc3 = __builtin_amdgcn_wmma_f32_16x16x4_f32(false, a, false, b, (short)0, acc3, false, false);
    }

#pragma unroll
    for (int j = 0; j < 8; ++j) {
        const long m = row0 + j + 8 * half;
        float* yp = Y + m * HID + r;
        yp[0]  = acc0[j];
        yp[16] = acc1[j];
        yp[32] = acc2[j];
        yp[48] = acc3[j];
    }
}

// ---------------------------------------------------------------------------
// agg[i,:] = xw[i,:] * dinv[i]^2     (self-loop term initializes accumulator)
// float4 vectorized: 64 % 4 == 0 so a float4 never crosses a node row.
// ---------------------------------------------------------------------------
__global__ void k_init_agg(const float* __restrict__ xw, const float* __restrict__ dinv,
                           float* __restrict__ agg) {
    long t    = (long)blockIdx.x * blockDim.x + threadIdx.x;
    long base = t * 4;
    int  node = (int)(base >> 6);
    float c = dinv[node];
    c = c * c;
    float4 v = *(const float4*)(xw + base);
    float4 o = make_float4(v.x * c, v.y * c, v.z * c, v.w * c);
    *(float4*)(agg + base) = o;
}

// ---------------------------------------------------------------------------
// Edge aggregation: agg[dst,:] += xw[src,:] * dinv[src]*dinv[dst]
// 32 lanes per edge, each lane handles 2 consecutive features (float2 gather,
// two f32 atomics). agg + xw both fit in the 192MB L2 -> atomics stay on-chip.
// ---------------------------------------------------------------------------
__global__ void k_edge_agg(const float* __restrict__ xw, const float* __restrict__ dinv,
                           const int* __restrict__ src, const int* __restrict__ dst,
                           float* agg) {
    long t = (long)blockIdx.x * blockDim.x + threadIdx.x;
    int  e = (int)(t >> 5);
    int  f = ((int)t & 31) * 2;
    int  s = src[e];
    int  d = dst[e];
    float c = dinv[s] * dinv[d];
    float2 v = *(const float2*)(xw + (long)s * HID + f);
    float* ap = agg + (long)d * HID + f;
    atomicAdd(ap,     v.x * c);
    atomicAdd(ap + 1, v.y * c);
}

// ---------------------------------------------------------------------------
// h = leaky_relu(agg + b)   (float4 vectorized)
// ---------------------------------------------------------------------------
__global__ void k_bias_lrelu(const float* __restrict__ agg, const float* __restrict__ b,
                             float* __restrict__ h) {
    long t    = (long)blockIdx.x * blockDim.x + threadIdx.x;
    long base = t * 4;
    int  f    = (int)(base & 63);
    float4 v  = *(const float4*)(agg + base);
    float4 bb = *(const float4*)(b + f);
    float4 o;
    o.x = v.x + bb.x; o.x = (o.x >= 0.0f) ? o.x : 0.2f * o.x;
    o.y = v.y + bb.y; o.y = (o.y >= 0.0f) ? o.y : 0.2f * o.y;
    o.z = v.z + bb.z; o.z = (o.z >= 0.0f) ? o.z : 0.2f * o.z;
    o.w = v.w + bb.w; o.w = (o.w >= 0.0f) ? o.w : 0.2f * o.w;
    *(float4*)(h + base) = o;
}

// ---------------------------------------------------------------------------
// Per-graph FC + sigmoid: out[g] = sigmoid( h[g,:] . fc_w + fc_b )
// One 256-thread block per graph (512*64 = 32768 elements), LDS reduction.
// ---------------------------------------------------------------------------
__global__ void k_fc_sigmoid(const float* __restrict__ h, const float* __restrict__ fcw,
                             const float* __restrict__ fcb, float* __restrict__ out) {
    __shared__ float red[256];
    const int g = blockIdx.x;
    const float* hp = h + (long)g * (N_NODES * HID);
    float sum = 0.0f;
    for (int j = threadIdx.x; j < N_NODES * HID; j += 256)
        sum += hp[j] * fcw[j];
    red[threadIdx.x] = sum;
    __syncthreads();
    for (int s = 128; s > 0; s >>= 1) {
        if (threadIdx.x < s) red[threadIdx.x] += red[threadIdx.x + s];
        __syncthreads();
    }
    if (threadIdx.x == 0) {
        float logit = red[0] + fcb[0];
        out[g] = 1.0f / (1.0f + expf(-logit));
    }
}

// ---------------------------------------------------------------------------
// Launch
// ---------------------------------------------------------------------------
extern "C" void kernel_launch(void* const* d_in, const int* in_sizes, int n_in,
                              void* d_out, int out_size, void* d_ws, size_t ws_size,
                              hipStream_t stream) {
    const float* x   = (const float*)d_in[0];
    const int*   el  = (const int*)d_in[1];
    const float* W1  = (const float*)d_in[2];
    const float* b1  = (const float*)d_in[3];
    const float* W2  = (const float*)d_in[4];
    const float* b2  = (const float*)d_in[5];
    const float* W3  = (const float*)d_in[6];
    const float* b3  = (const float*)d_in[7];
    const float* fcw = (const float*)d_in[8];
    const float* fcb = (const float*)d_in[9];
    float* out = (float*)d_out;

    const int* src = el;
    const int* dst = el + NEDGE;

    char*  ws   = (char*)d_ws;
    float* P    = (float*)ws;                                  // xw / h   (33.5 MB)
    float* Q    = (float*)(ws + (size_t)NTOT * HID * 4);       // agg      (33.5 MB)
    float* dinv = (float*)(ws + (size_t)NTOT * HID * 8);       // rsqrt(deg) (0.5 MB)

    const int THR = 256;
    const int gN      = NTOT / THR;            // 512
    const int gE      = NEDGE / THR;           // 8192
    const int gE32    = (int)((long)NEDGE * 32 / THR);   // 262144
    const int gElem4  = (int)((long)NTOT * HID / 4 / THR); // 8192
    const int gGemm   = NTOT / (16 * (THR / 32));        // 1024 (8 waves * 16 rows)

    // Degrees -> dinv
    k_fill_ones <<<gN,  THR, 0, stream>>>(dinv);
    k_deg_count <<<gE,  THR, 0, stream>>>(dst, dinv);
    k_deg_rsqrt <<<gN,  THR, 0, stream>>>(dinv);

    // Layer 1 (K = 32)
    k_gemm_wmma<FEAT><<<gGemm, THR, 0, stream>>>(x, W1, P);
    k_init_agg  <<<gElem4, THR, 0, stream>>>(P, dinv, Q);
    k_edge_agg  <<<gE32,   THR, 0, stream>>>(P, dinv, src, dst, Q);
    k_bias_lrelu<<<gElem4, THR, 0, stream>>>(Q, b1, P);

    // Layer 2 (K = 64, GEMM in place on P)
    k_gemm_wmma<HID><<<gGemm, THR, 0, stream>>>(P, W2, P);
    k_init_agg  <<<gElem4, THR, 0, stream>>>(P, dinv, Q);
    k_edge_agg  <<<gE32,   THR, 0, stream>>>(P, dinv, src, dst, Q);
    k_bias_lrelu<<<gElem4, THR, 0, stream>>>(Q, b2, P);

    // Layer 3 (K = 64)
    k_gemm_wmma<HID><<<gGemm, THR, 0, stream>>>(P, W3, P);
    k_init_agg  <<<gElem4, THR, 0, stream>>>(P, dinv, Q);
    k_edge_agg  <<<gE32,   THR, 0, stream>>>(P, dinv, src, dst, Q);
    k_bias_lrelu<<<gElem4, THR, 0, stream>>>(Q, b3, P);

    // Readout
    k_fc_sigmoid<<<BATCH, THR, 0, stream>>>(P, fcw, fcb, out);
}